// GCNModel_16999480557923
// MI455X (gfx1250) — compile-verified
//
#include <hip/hip_runtime.h>
#include <hip/hip_bf16.h>

typedef __attribute__((ext_vector_type(2))) float v2f;
typedef __attribute__((ext_vector_type(8))) float v8f;

// ---------------------------------------------------------------------------
// Utility: zero a float buffer (accumulators must be re-zeroed every call).
// ---------------------------------------------------------------------------
__global__ void zero_f32(float* __restrict__ p, long long n) {
    long long i  = (long long)blockIdx.x * blockDim.x + threadIdx.x;
    long long st = (long long)gridDim.x * blockDim.x;
    for (; i < n; i += st) p[i] = 0.0f;
}

// ---------------------------------------------------------------------------
// Degree count (float atomics; counts < 2^24 so exact) and dinv = rsqrt(deg+1)
// ---------------------------------------------------------------------------
__global__ void deg_count(const long long* __restrict__ dst, long long E,
                          float* __restrict__ deg) {
    long long i  = (long long)blockIdx.x * blockDim.x + threadIdx.x;
    long long st = (long long)gridDim.x * blockDim.x;
    for (; i < E; i += st) atomicAdd(&deg[dst[i]], 1.0f);
}

__global__ void make_dinv(const float* __restrict__ deg, float* __restrict__ dinv, int n) {
    int i = blockIdx.x * blockDim.x + threadIdx.x;
    if (i < n) dinv[i] = rsqrtf(deg[i] + 1.0f);  // +1 for self-loop; always > 0
}

// ---------------------------------------------------------------------------
// GEMM Y[M x 128] = X[M x 128] @ W[128 x 128] via v_wmma_f32_16x16x4_f32.
// 256 threads = 8 waves; block owns one 16-row M tile; wave w owns N tile w.
// Fragment layouts per CDNA5 ISA 7.12.2 (32-bit A 16x4, B/C/D row-striped).
// ---------------------------------------------------------------------------
__global__ void __launch_bounds__(256) gemm_xw_128(const float* __restrict__ X,
                                                   const float* __restrict__ W,
                                                   float* __restrict__ Y, int M) {
    const int K = 128, Nf = 128;
    const int wave = threadIdx.x >> 5;
    const int lane = threadIdx.x & 31;
    const int half = lane >> 4;        // 0: lanes 0-15, 1: lanes 16-31
    const int lrow = lane & 15;
    const int m0 = blockIdx.x * 16;
    const int n0 = wave * 16;

    int arow = m0 + lrow;
    if (arow >= M) arow = M - 1;       // clamp pad rows (stores are guarded)
    const float* xrow = X + (long long)arow * K;

    v8f acc = {};
    #pragma unroll
    for (int k0 = 0; k0 < K; k0 += 4) {
        const int ka = k0 + 2 * half;
        v2f a; a.x = xrow[ka];                       a.y = xrow[ka + 1];
        v2f b; b.x = W[(long long)ka * Nf + n0 + lrow];
               b.y = W[(long long)(ka + 1) * Nf + n0 + lrow];
        acc = __builtin_amdgcn_wmma_f32_16x16x4_f32(
            false, a, false, b, (short)0, acc, false, false);
    }
    #pragma unroll
    for (int i = 0; i < 8; ++i) {
        const int m = m0 + i + 8 * half;
        if (m < M) Y[(long long)m * Nf + n0 + lrow] = acc[i];
    }
}

// ---------------------------------------------------------------------------
// GEMM Y[M x 16] = X[M x 128] @ W[128 x 16]; wave w owns M sub-tile w.
// ---------------------------------------------------------------------------
__global__ void __launch_bounds__(256) gemm_xw_16(const float* __restrict__ X,
                                                  const float* __restrict__ W,
                                                  float* __restrict__ Y, int M) {
    const int K = 128, Nf = 16;
    const int wave = threadIdx.x >> 5;
    const int lane = threadIdx.x & 31;
    const int half = lane >> 4;
    const int lrow = lane & 15;
    const int m0 = (blockIdx.x * 8 + wave) * 16;
    if (m0 >= M) return;

    int arow = m0 + lrow;
    if (arow >= M) arow = M - 1;
    const float* xrow = X + (long long)arow * K;

    v8f acc = {};
    #pragma unroll
    for (int k0 = 0; k0 < K; k0 += 4) {
        const int ka = k0 + 2 * half;
        v2f a; a.x = xrow[ka];             a.y = xrow[ka + 1];
        v2f b; b.x = W[ka * Nf + lrow];    b.y = W[(ka + 1) * Nf + lrow];
        acc = __builtin_amdgcn_wmma_f32_16x16x4_f32(
            false, a, false, b, (short)0, acc, false, false);
    }
    #pragma unroll
    for (int i = 0; i < 8; ++i) {
        const int m = m0 + i + 8 * half;
        if (m < M) Y[(long long)m * Nf + lrow] = acc[i];
    }
}

// ---------------------------------------------------------------------------
// Edge aggregation, F = 128: one wave per edge, float4 per lane.
// xt (51 MB) and agg (51 MB) are L2-resident on MI455X (192 MB L2).
// ---------------------------------------------------------------------------
__global__ void __launch_bounds__(256) edge_agg128(const float* __restrict__ xt,
                                                   const float* __restrict__ dinv,
                                                   const long long* __restrict__ src,
                                                   const long long* __restrict__ dst,
                                                   float* __restrict__ agg, long long E) {
    const int lane = threadIdx.x & 31;
    long long wave = ((long long)blockIdx.x * blockDim.x + threadIdx.x) >> 5;
    long long nw   = ((long long)gridDim.x * blockDim.x) >> 5;
    for (long long e = wave; e < E; e += nw) {
        const long long s = src[e];
        const long long d = dst[e];
        const float w = dinv[s] * dinv[d];
        const float4 v = *(const float4*)(xt + s * 128 + lane * 4);
        float* o = agg + d * 128 + lane * 4;
        atomicAdd(o + 0, v.x * w);
        atomicAdd(o + 1, v.y * w);
        atomicAdd(o + 2, v.z * w);
        atomicAdd(o + 3, v.w * w);
    }
}

// ---------------------------------------------------------------------------
// Edge aggregation, F = 16: 4 lanes per edge (8 edges per wave).
// ---------------------------------------------------------------------------
__global__ void __launch_bounds__(256) edge_agg16(const float* __restrict__ xt,
                                                  const float* __restrict__ dinv,
                                                  const long long* __restrict__ src,
                                                  const long long* __restrict__ dst,
                                                  float* __restrict__ agg, long long E) {
    const int lane = threadIdx.x & 31;
    const int sub  = lane >> 2;          // 0..7 : edge within the wave's group
    const int fb   = (lane & 3) * 4;     // feature base: 0,4,8,12
    long long wave = ((long long)blockIdx.x * blockDim.x + threadIdx.x) >> 5;
    long long nw   = ((long long)gridDim.x * blockDim.x) >> 5;
    for (long long eb = wave * 8; eb < E; eb += nw * 8) {
        const long long e = eb + sub;
        if (e < E) {
            const long long s = src[e];
            const long long d = dst[e];
            const float w = dinv[s] * dinv[d];
            const float4 v = *(const float4*)(xt + s * 16 + fb);
            float* o = agg + d * 16 + fb;
            atomicAdd(o + 0, v.x * w);
            atomicAdd(o + 1, v.y * w);
            atomicAdd(o + 2, v.z * w);
            atomicAdd(o + 3, v.w * w);
        }
    }
}

// ---------------------------------------------------------------------------
// Layer-1 epilogue: h = relu(agg + xt*dinv^2 + b), in place over agg (F=128).
// ---------------------------------------------------------------------------
__global__ void finish1(float* __restrict__ h, const float* __restrict__ xt,
                        const float* __restrict__ dinv, const float* __restrict__ b,
                        long long total) {
    long long i  = (long long)blockIdx.x * blockDim.x + threadIdx.x;
    long long st = (long long)gridDim.x * blockDim.x;
    for (; i < total; i += st) {
        const long long node = i >> 7;
        const int f = (int)(i & 127);
        const float di = dinv[node];
        const float v = h[i] + xt[i] * di * di + b[f];
        h[i] = v > 0.0f ? v : 0.0f;
    }
}

// ---------------------------------------------------------------------------
// Layer-2 epilogue: out += xt*dinv^2 + b (F=16), no ReLU.
// ---------------------------------------------------------------------------
__global__ void finish2(float* __restrict__ out, const float* __restrict__ xt,
                        const float* __restrict__ dinv, const float* __restrict__ b,
                        long long total) {
    long long i  = (long long)blockIdx.x * blockDim.x + threadIdx.x;
    long long st = (long long)gridDim.x * blockDim.x;
    for (; i < total; i += st) {
        const long long node = i >> 4;
        const int f = (int)(i & 15);
        const float di = dinv[node];
        out[i] = out[i] + xt[i] * di * di + b[f];
    }
}

// ---------------------------------------------------------------------------
// Launch: x, edge_index(int64 [2,E]), W1, b1, W2, b2  ->  out [N,16] f32
// ---------------------------------------------------------------------------
extern "C" void kernel_launch(void* const* d_in, const int* in_sizes, int n_in,
                              void* d_out, int out_size, void* d_ws, size_t ws_size,
                              hipStream_t stream) {
    const float*     x   = (const float*)d_in[0];
    const long long* ei  = (const long long*)d_in[1];
    const float*     W1  = (const float*)d_in[2];
    const float*     b1  = (const float*)d_in[3];
    const float*     W2  = (const float*)d_in[4];
    const float*     b2  = (const float*)d_in[5];
    float*           out = (float*)d_out;

    const int       Nn = in_sizes[0] / 128;         // 100000 nodes
    const long long E  = (long long)in_sizes[1] / 2; // 1.6M edges
    const long long* src = ei;
    const long long* dst = ei + E;

    float* ws   = (float*)d_ws;
    float* xt1  = ws;                                  // Nn*128
    float* h    = xt1 + (long long)Nn * 128;           // Nn*128 (agg1, then h)
    float* xt2  = h   + (long long)Nn * 128;           // Nn*16
    float* deg  = xt2 + (long long)Nn * 16;            // Nn
    float* dinv = deg + Nn;                            // Nn

    // --- normalization ---
    zero_f32<<<512, 256, 0, stream>>>(deg, (long long)Nn);
    zero_f32<<<4096, 256, 0, stream>>>(h, (long long)Nn * 128);
    zero_f32<<<1024, 256, 0, stream>>>(out, (long long)Nn * 16);
    deg_count<<<2048, 256, 0, stream>>>(dst, E, deg);
    make_dinv<<<(Nn + 255) / 256, 256, 0, stream>>>(deg, dinv, Nn);

    // --- layer 1 ---
    gemm_xw_128<<<(Nn + 15) / 16, 256, 0, stream>>>(x, W1, xt1, Nn);
    edge_agg128<<<8192, 256, 0, stream>>>(xt1, dinv, src, dst, h, E);
    finish1<<<4096, 256, 0, stream>>>(h, xt1, dinv, b1, (long long)Nn * 128);

    // --- layer 2 ---
    gemm_xw_16<<<(Nn + 127) / 128, 256, 0, stream>>>(h, W2, xt2, Nn);
    edge_agg16<<<4096, 256, 0, stream>>>(xt2, dinv, src, dst, out, E);
    finish2<<<2048, 256, 0, stream>>>(out, xt2, dinv, b2, (long long)Nn * 16);
}